// PagedAttentionOp_22497038697045
// MI455X (gfx1250) — compile-verified
//
#include <hip/hip_runtime.h>
#include <float.h>

// ---------------------------------------------------------------------------
// Paged attention decode (q_len == 1), fp32, GQA 32q/8kv, D=128, pages of 16.
// CDNA5 / gfx1250: wave32, V_WMMA_F32_16X16X4_F32 for both QK^T and PV.
// ---------------------------------------------------------------------------

typedef __attribute__((ext_vector_type(2))) float v2f;
typedef __attribute__((ext_vector_type(8))) float v8f;

#define B_DIM     16
#define QH_DIM    32
#define KVH_DIM   8
#define GQA       4      // QH / KVH
#define HD        128    // head dim
#define NPAGES    4096
#define TPP       16     // tokens per page (== WMMA M tile)
#define MAX_PPS   256

#define NW        8              // waves per block (each wave = one seq chunk)
#define NTHREADS  (NW * 32)
#define BK        (B_DIM * KVH_DIM)   // 128 (b, kvh) pairs
#define NSPLIT    4              // block-level sequence splits (2-pass path)

static __device__ __forceinline__ v8f wmma4(v2f a, v2f b, v8f c) {
  // D(16x16,f32) = A(16x4,f32) * B(4x16,f32) + C
  return __builtin_amdgcn_wmma_f32_16x16x4_f32(false, a, false, b,
                                               (short)0, c, false, false);
}

// ---------------------------------------------------------------------------
// Pass 1: each wave flash-attends over its chunk of pages; waves combine via
// LDS. FINAL=true writes the output directly (nsplit==1); FINAL=false writes
// unnormalized partials (acc, m, l) to workspace for pass 2.
// ---------------------------------------------------------------------------
template <bool FINAL>
__global__ __launch_bounds__(NTHREADS)
void pa_pass1(const float* __restrict__ q,        // [B, 1, QH, HD]
              const float* __restrict__ kpages,   // [KVH, NPAGES, TPP, HD]
              const float* __restrict__ vpages,   // [KVH, NPAGES, TPP, HD]
              const int*   __restrict__ seq_len,  // [B]
              const int*   __restrict__ page_map, // [B, MAX_PPS]
              float* __restrict__ out,            // [B, QH, HD]     (FINAL)
              float* __restrict__ ws_acc,         // [BK, nsplit, GQA*HD]
              float* __restrict__ ws_m,           // [BK, nsplit, GQA]
              float* __restrict__ ws_l,           // [BK, nsplit, GQA]
              int nsplit)
{
  const int blk   = blockIdx.x;
  const int split = blk % nsplit;
  const int bk    = blk / nsplit;
  const int b     = bk / KVH_DIM;
  const int kvh   = bk % KVH_DIM;

  const int tid  = threadIdx.x;
  const int w    = tid >> 5;     // wave id in block
  const int lane = tid & 31;
  const int h    = lane & 15;    // head column (valid heads: 0..3)
  const int hl   = lane >> 4;    // lane half

  const int len          = seq_len[b];
  const int pages_needed = (len + TPP - 1) / TPP;
  const int ppw          = MAX_PPS / (nsplit * NW);   // pages per wave-chunk
  const int wchunk       = split * NW + w;
  const int p0           = wchunk * ppw;
  const int p1           = min(p0 + ppw, pages_needed);

  // --- Preload Q in WMMA B-operand layout (reused for every page) ---------
  // B(4x16): vgpr v, lane L -> k = 2*(L/16)+v (d within chunk), n = L%16 (head)
  v2f qB[32];
  {
    const float* qrow =
        q + ((size_t)b * QH_DIM + kvh * GQA + (h < GQA ? h : 0)) * HD;
    #pragma unroll
    for (int c = 0; c < 32; ++c) {
      v2f t = *(const v2f*)(qrow + 4 * c + 2 * hl);
      qB[c].x = (h < GQA) ? t.x : 0.f;
      qB[c].y = (h < GQA) ? t.y : 0.f;
    }
  }

  // out^T accumulators: 8 M-tiles of 16 d-columns; C layout:
  // vgpr r, lane L -> d = 16*mt + r + 8*(L/16), head = L%16
  v8f acc[8];
  #pragma unroll
  for (int mt = 0; mt < 8; ++mt)
    #pragma unroll
    for (int r = 0; r < 8; ++r) acc[mt][r] = 0.f;

  float m_run = -FLT_MAX;
  float l_run = 0.f;

  const int*   pmap     = page_map + (size_t)b * MAX_PPS;
  const size_t head_off = (size_t)kvh * NPAGES * TPP * HD;

  for (int p = p0; p < p1; ++p) {
    const int    pid = pmap[p];
    const float* kb  = kpages + head_off + (size_t)pid * (TPP * HD);
    const float* vb  = vpages + head_off + (size_t)pid * (TPP * HD);

    // Prefetch next page (whole 8KB K and V pages, 256B/lane).
    if (p + 1 < p1) {
      const int pid2 = pmap[p + 1];
      __builtin_prefetch(kpages + head_off + (size_t)pid2 * (TPP * HD) + lane * 64, 0, 0);
      __builtin_prefetch(vpages + head_off + (size_t)pid2 * (TPP * HD) + lane * 64, 0, 0);
    }

    // --- S^T = K * Q^T : 16 tokens x 16 heads (4 live) -------------------
    // A(16x4): lane L -> m = token L%16, k = d = 4c + 2*(L/16) + {0,1}
    v8f sC0, sC1;
    #pragma unroll
    for (int r = 0; r < 8; ++r) { sC0[r] = 0.f; sC1[r] = 0.f; }
    const float* krow = kb + h * HD + 2 * hl;
    #pragma unroll
    for (int c = 0; c < 32; c += 2) {
      v2f a0 = *(const v2f*)(krow + 4 * c);
      v2f a1 = *(const v2f*)(krow + 4 * (c + 1));
      sC0 = wmma4(a0, qB[c], sC0);
      sC1 = wmma4(a1, qB[c + 1], sC1);
    }

    // --- mask + online softmax (tokens live across vgprs + lane halves) --
    const int tokbase = p * TPP + 8 * hl;
    float s0[8], P[8];
    float tmax = -FLT_MAX;
    #pragma unroll
    for (int r = 0; r < 8; ++r) {
      float sv = sC0[r] + sC1[r];
      sv = (tokbase + r < len) ? sv : -FLT_MAX;   // matches finfo.min mask
      s0[r] = sv;
      tmax = fmaxf(tmax, sv);
    }
    tmax = fmaxf(tmax, __shfl_xor(tmax, 16, 32));
    const float m_new = fmaxf(m_run, tmax);
    const float resc  = __expf(m_run - m_new);
    float tsum = 0.f;
    #pragma unroll
    for (int r = 0; r < 8; ++r) {
      P[r] = __expf(s0[r] - m_new);
      tsum += P[r];
    }
    tsum += __shfl_xor(tsum, 16, 32);
    l_run = l_run * resc + tsum;
    m_run = m_new;
    #pragma unroll
    for (int mt = 0; mt < 8; ++mt)
      #pragma unroll
      for (int r = 0; r < 8; ++r) acc[mt][r] *= resc;

    // Cross-half copies of P to build P^T in B-operand layout.
    float sw[8];
    #pragma unroll
    for (int r = 0; r < 8; ++r) sw[r] = __shfl_xor(P[r], 16, 32);
    const bool half0 = (hl == 0);

    // --- out^T += V^T * P^T : 8 M-tiles (d) x 4 K-steps (token groups) ---
    #pragma unroll
    for (int s = 0; s < 4; ++s) {
      v2f bP;  // B(4x16): k = token 4s+2*hl+v, n = head L%16
      if (s == 0) { bP.x = half0 ? P[0]  : sw[2]; bP.y = half0 ? P[1]  : sw[3]; }
      if (s == 1) { bP.x = half0 ? P[4]  : sw[6]; bP.y = half0 ? P[5]  : sw[7]; }
      if (s == 2) { bP.x = half0 ? sw[0] : P[2];  bP.y = half0 ? sw[1] : P[3];  }
      if (s == 3) { bP.x = half0 ? sw[4] : P[6];  bP.y = half0 ? sw[5] : P[7];  }
      const float* vrow = vb + (4 * s + 2 * hl) * HD + h;  // V[t0][16mt + h]
      #pragma unroll
      for (int mt = 0; mt < 8; ++mt) {
        v2f a;                     // A(16x4) = V^T tile: m = d, k = token
        a.x = vrow[mt * 16];
        a.y = vrow[mt * 16 + HD];
        acc[mt] = wmma4(a, bP, acc[mt]);
      }
    }
  }

  // --- combine the NW waves of this block via LDS -------------------------
  __shared__ float lds_m[NW][GQA];
  __shared__ float lds_l[NW][GQA];
  __shared__ float lds_acc[NW][GQA * HD];

  if (h < GQA) {
    if (hl == 0) { lds_m[w][h] = m_run; lds_l[w][h] = l_run; }
    #pragma unroll
    for (int mt = 0; mt < 8; ++mt)
      #pragma unroll
      for (int r = 0; r < 8; ++r)
        lds_acc[w][h * HD + mt * 16 + 8 * hl + r] = acc[mt][r];
  }
  __syncthreads();

  for (int idx = tid; idx < GQA * HD; idx += NTHREADS) {
    const int g = idx >> 7;
    const int d = idx & 127;
    float mg = -FLT_MAX;
    #pragma unroll
    for (int ww = 0; ww < NW; ++ww) mg = fmaxf(mg, lds_m[ww][g]);
    float lg = 0.f, ag = 0.f;
    #pragma unroll
    for (int ww = 0; ww < NW; ++ww) {
      const float e = __expf(lds_m[ww][g] - mg);
      lg += lds_l[ww][g] * e;
      ag += lds_acc[ww][g * HD + d] * e;
    }
    if (FINAL) {
      out[((size_t)b * QH_DIM + kvh * GQA + g) * HD + d] = ag / lg;
    } else {
      const size_t pb = (size_t)(bk * nsplit + split);
      ws_acc[pb * (GQA * HD) + idx] = ag;
      if (d == 0) { ws_m[pb * GQA + g] = mg; ws_l[pb * GQA + g] = lg; }
    }
  }
}

// ---------------------------------------------------------------------------
// Pass 2: merge the block-level sequence splits (tiny: 128 blocks x 512 thr).
// ---------------------------------------------------------------------------
__global__ __launch_bounds__(GQA * HD)
void pa_pass2(const float* __restrict__ ws_acc,
              const float* __restrict__ ws_m,
              const float* __restrict__ ws_l,
              float* __restrict__ out, int nsplit)
{
  const int bk  = blockIdx.x;
  const int idx = threadIdx.x;     // 0..511 = g*128 + d
  const int g   = idx >> 7;
  const int d   = idx & 127;

  float mg = -FLT_MAX;
  for (int s = 0; s < nsplit; ++s)
    mg = fmaxf(mg, ws_m[(size_t)(bk * nsplit + s) * GQA + g]);
  float lg = 0.f, ag = 0.f;
  for (int s = 0; s < nsplit; ++s) {
    const size_t pb = (size_t)(bk * nsplit + s);
    const float  e  = __expf(ws_m[pb * GQA + g] - mg);
    lg += ws_l[pb * GQA + g] * e;
    ag += ws_acc[pb * (GQA * HD) + idx] * e;
  }
  const int b = bk / KVH_DIM, kvh = bk % KVH_DIM;
  out[((size_t)b * QH_DIM + kvh * GQA + g) * HD + d] = ag / lg;
}

// ---------------------------------------------------------------------------
extern "C" void kernel_launch(void* const* d_in, const int* in_sizes, int n_in,
                              void* d_out, int out_size, void* d_ws,
                              size_t ws_size, hipStream_t stream) {
  (void)in_sizes; (void)n_in; (void)out_size;
  const float* q  = (const float*)d_in[0];
  const float* kp = (const float*)d_in[1];
  const float* vp = (const float*)d_in[2];
  const int*   sl = (const int*)d_in[3];
  const int*   pm = (const int*)d_in[4];
  float*       out = (float*)d_out;

  const size_t need =
      (size_t)BK * NSPLIT * (GQA * HD + 2 * GQA) * sizeof(float);

  if (ws_size >= need) {
    // 2-pass flash-decode: 512 blocks x 8 waves in pass 1.
    float* ws_acc = (float*)d_ws;
    float* ws_m   = ws_acc + (size_t)BK * NSPLIT * GQA * HD;
    float* ws_l   = ws_m + (size_t)BK * NSPLIT * GQA;
    pa_pass1<false><<<BK * NSPLIT, NTHREADS, 0, stream>>>(
        q, kp, vp, sl, pm, nullptr, ws_acc, ws_m, ws_l, NSPLIT);
    pa_pass2<<<BK, GQA * HD, 0, stream>>>(ws_acc, ws_m, ws_l, out, NSPLIT);
  } else {
    // Fallback: single pass, waves-of-block combine in LDS only.
    pa_pass1<true><<<BK, NTHREADS, 0, stream>>>(
        q, kp, vp, sl, pm, out, nullptr, nullptr, nullptr, 1);
  }
}